// NuFFT_25108378812741
// MI455X (gfx1250) — compile-verified
//
#include <hip/hip_runtime.h>
#include <math.h>

// ---------------- problem constants ----------------
#define NPIX   1024
#define NCHAN  4
#define NVIS   200000
#define OSF    2
#define GRIDN  2048           // NPIX * OSF
#define KW     6              // kernel width W
#define PI_F   3.14159265358979323846f
#define BETA_F 13.8551018f    // pi * sqrt((W/OSF)^2*(OSF-0.5)^2 - 0.8) = pi*sqrt(19.45)
// g = u * 1000 * DL * GRID ; DL = 0.005 * pi/(180*3600)
#define GSCALE 4.96449110e-2f

// ---- precomputed-table layout (floats), shared between global and LDS ----
// D32 planes (re, im, -im), 32 rows x pitch 36
#define D32_PITCH  36
#define D32_PLANE  (32 * D32_PITCH)        // 1152
#define OFF_D32RE  0
#define OFF_D32IM  (OFF_D32RE + D32_PLANE)
#define OFF_D32NI  (OFF_D32IM + D32_PLANE)
// D64 planes (re, im, -im), 64 rows x pitch 68
#define D64_PITCH  68
#define D64_PLANE  (64 * D64_PITCH)        // 4352
#define OFF_D64RE  (OFF_D32NI + D32_PLANE) // 3456
#define OFF_D64IM  (OFF_D64RE + D64_PLANE)
#define OFF_D64NI  (OFF_D64IM + D64_PLANE)
// twiddle table T[n1][k2] = exp(-2pi i n1 k2 / 2048), 64 rows x pitch 36
#define TW_PITCH   36
#define TW_PLANE   (64 * TW_PITCH)         // 2304
#define OFF_TWRE   (OFF_D64NI + D64_PLANE) // 16512
#define OFF_TWIM   (OFF_TWRE + TW_PLANE)
#define TBL_FLOATS (OFF_TWIM + TW_PLANE)   // 21120 floats = 84480 B (16B multiple)

// ---- CDNA5 async global->LDS (ASYNCcnt) with safe fallback ----
#if __has_builtin(__builtin_amdgcn_global_load_async_to_lds_b128)
#define HAVE_ASYNC_LDS 1
typedef int v4i_t __attribute__((ext_vector_type(4)));
typedef __attribute__((address_space(1))) v4i_t* gv4i_p;   // global (AS1)
typedef __attribute__((address_space(3))) v4i_t* lv4i_p;   // LDS    (AS3)
__device__ __forceinline__ void async_cp16(const float* gsrc, float* ldst) {
    __builtin_amdgcn_global_load_async_to_lds_b128((gv4i_p)gsrc, (lv4i_p)ldst, 0, 0);
}
__device__ __forceinline__ void async_wait0() {
#if __has_builtin(__builtin_amdgcn_s_wait_asynccnt)
    __builtin_amdgcn_s_wait_asynccnt(0);
#else
    asm volatile("s_wait_asynccnt 0" ::: "memory");
#endif
}
#else
#define HAVE_ASYNC_LDS 0
#endif

// ---------------- WMMA f32 16x16x4 helpers ----------------
typedef float v2f __attribute__((ext_vector_type(2)));
typedef float v8f __attribute__((ext_vector_type(8)));

// A fragment: 16x4 (MxK). lanes 0-15 -> M=0..15 ; VGPR0: K = 2*(lane>=16), VGPR1: K+1
__device__ __forceinline__ v2f frag_a(const float* m, int pitch, int mbase, int kbase, int lane) {
    int row = mbase + (lane & 15);
    int k0  = kbase + ((lane >> 4) << 1);
    v2f a;
    a.x = m[row * pitch + k0];
    a.y = m[row * pitch + k0 + 1];
    return a;
}
// B fragment: 4x16 (KxN). lanes 0-15 -> N=0..15 ; VGPR0: K = 2*(lane>=16), VGPR1: K+1
__device__ __forceinline__ v2f frag_b(const float* m, int pitch, int kbase, int nbase, int lane) {
    int col = nbase + (lane & 15);
    int k0  = kbase + ((lane >> 4) << 1);
    v2f b;
    b.x = m[k0 * pitch + col];
    b.y = m[(k0 + 1) * pitch + col];
    return b;
}
// C/D fragment: 16x16, VGPR r -> M = r + 8*(lane>=16), N = lane&15
__device__ __forceinline__ void store_c(float* m, int pitch, int mbase, int nbase, int lane, v8f c) {
    int col  = nbase + (lane & 15);
    int half = (lane >> 4) << 3;
#pragma unroll
    for (int r = 0; r < 8; ++r)
        m[(mbase + half + r) * pitch + col] = c[r];
}

// Complex tile matmul: one 16x16 output tile of C = A * B (complex).
// f32 WMMA has no A/B negation (NEG = CNeg only), so A comes as (Re, Im, -Im):
// Cre = Are*Bre + (-Aim)*Bim ; Cim = Are*Bim + Aim*Bre.
__device__ __forceinline__ void cmatmul_tile(
    const float* Are, const float* Aim, const float* Ani, int ap,
    const float* Bre, const float* Bim, int bp,
    float* Cre, float* Cim, int cp,
    int mbase, int nbase, int K, int lane) {
    v8f accRe = {0.f,0.f,0.f,0.f,0.f,0.f,0.f,0.f};
    v8f accIm = {0.f,0.f,0.f,0.f,0.f,0.f,0.f,0.f};
    for (int kb = 0; kb < K; kb += 4) {
        v2f are = frag_a(Are, ap, mbase, kb, lane);
        v2f aim = frag_a(Aim, ap, mbase, kb, lane);
        v2f ani = frag_a(Ani, ap, mbase, kb, lane);
        v2f bre = frag_b(Bre, bp, kb, nbase, lane);
        v2f bim = frag_b(Bim, bp, kb, nbase, lane);
        accRe = __builtin_amdgcn_wmma_f32_16x16x4_f32(false, are, false, bre, (short)0, accRe, false, false);
        accRe = __builtin_amdgcn_wmma_f32_16x16x4_f32(false, ani, false, bim, (short)0, accRe, false, false);
        accIm = __builtin_amdgcn_wmma_f32_16x16x4_f32(false, are, false, bim, (short)0, accIm, false, false);
        accIm = __builtin_amdgcn_wmma_f32_16x16x4_f32(false, aim, false, bre, (short)0, accIm, false, false);
    }
    store_c(Cre, cp, mbase, nbase, lane, accRe);
    store_c(Cim, cp, mbase, nbase, lane, accIm);
}

// ---------------- kernel 0: build DFT/twiddle tables once ----------------
__global__ void nufft_make_tables(float* __restrict__ tbl) {
    int i = blockIdx.x * blockDim.x + threadIdx.x;
    if (i < D32_PLANE) {                    // DFT-32 matrix planes
        int r = i / D32_PITCH, q = i % D32_PITCH;
        float sn = 0.f, cs = 0.f;
        if (q < 32) {
            float ang = -2.f * PI_F * (float)((r * q) & 31) / 32.f;
            __sincosf(ang, &sn, &cs);
        }
        tbl[OFF_D32RE + i] = cs;
        tbl[OFF_D32IM + i] = sn;
        tbl[OFF_D32NI + i] = -sn;
    }
    if (i < D64_PLANE) {                    // DFT-64 matrix planes
        int r = i / D64_PITCH, q = i % D64_PITCH;
        float sn = 0.f, cs = 0.f;
        if (q < 64) {
            float ang = -2.f * PI_F * (float)((r * q) & 63) / 64.f;
            __sincosf(ang, &sn, &cs);
        }
        tbl[OFF_D64RE + i] = cs;
        tbl[OFF_D64IM + i] = sn;
        tbl[OFF_D64NI + i] = -sn;
    }
    if (i < TW_PLANE) {                     // four-step twiddles
        int n1 = i / TW_PITCH, k2 = i % TW_PITCH;
        float sn = 0.f, cs = 0.f;
        if (k2 < 32) {
            float ang = -2.f * PI_F * (float)(n1 * k2) / 2048.f;
            __sincosf(ang, &sn, &cs);
        }
        tbl[OFF_TWRE + i] = cs;
        tbl[OFF_TWIM + i] = sn;
    }
}

// ---------------- apodization ----------------
__device__ __forceinline__ float apod1d(int n) {
    float nn = (float)(n - NPIX / 2);
    float x  = PI_F * (float)KW * nn / (float)GRIDN;
    float s  = sqrtf(BETA_F * BETA_F - x * x);
    return (float)KW * sinhf(s) / s;
}

// ---------------- kernel 1: pad + roll + apod-correct ----------------
__global__ void nufft_prep(const float* __restrict__ img, float2* __restrict__ grid) {
    size_t idx = (size_t)blockIdx.x * blockDim.x + threadIdx.x;
    const size_t total = (size_t)NCHAN * GRIDN * GRIDN;
    if (idx >= total) return;
    int x = (int)(idx & (GRIDN - 1));
    int y = (int)((idx >> 11) & (GRIDN - 1));
    int c = (int)(idx >> 22);
    int ys = (y + NPIX / 2) & (GRIDN - 1);
    int xs = (x + NPIX / 2) & (GRIDN - 1);
    float2 v = make_float2(0.f, 0.f);
    if (ys < NPIX && xs < NPIX) {
        v.x = img[((size_t)c * NPIX + ys) * NPIX + xs] / (apod1d(ys) * apod1d(xs));
    }
    grid[idx] = v;
}

// ---------------- kernel 2: 2048-pt FFT along rows (four-step, WMMA) ------
// One workgroup (256 thr = 8 wave32) per row. Tables + row are staged into
// LDS with GLOBAL_LOAD_ASYNC_TO_LDS_B128 when available. Writes transposed
// so the second launch FFTs the other axis with identical code.
// LDS plan (floats), 39040 total = 156.2 KB (2 workgroups fit in 320 KB WGP):
//   [0,21120)          table mirror
//   [21120, +2176)x2   Xre, Xim   (32 x pitch 68)
//   [25472, +2176)x2   Zre, Zim   (32 x pitch 68)   <- aliased by Xint (4096) early
//   [29824, +2304)x2   Tre, Tim   (64 x pitch 36)
//   [34432, +2304)x2   Ore, Oim   (64 x pitch 36)
#define SM_X    (TBL_FLOATS)
#define SM_Z    (SM_X + 2 * 2176)
#define SM_T    (SM_Z + 2 * 2176)
#define SM_O    (SM_T + 2 * 2304)
#define SM_TOT  (SM_O + 2 * 2304)

__global__ __launch_bounds__(256) void nufft_fft2048(const float2* __restrict__ src,
                                                     float2* __restrict__ dst,
                                                     const float* __restrict__ gtbl) {
    const int row  = blockIdx.x;     // 0..GRIDN-1
    const int c    = blockIdx.y;     // channel
    const int tid  = threadIdx.x;
    const int lane = tid & 31;
    const int wave = tid >> 5;       // 0..7

    __shared__ __align__(16) float smem[SM_TOT];
    float* tbl  = smem;
    float* Xre  = smem + SM_X;
    float* Xim  = Xre + 2176;
    float* Zre  = smem + SM_Z;
    float* Zim  = Zre + 2176;
    float* Tre  = smem + SM_T;
    float* Tim  = Tre + 2304;
    float* Ore  = smem + SM_O;
    float* Oim  = Ore + 2304;
    float* Xint = Zre;               // interleaved row staging, dead before Z written

    const float* rowg = (const float*)(src + ((size_t)c * GRIDN + row) * GRIDN); // 4096 floats

#if HAVE_ASYNC_LDS
    // async-stage table (5280 x 16B) and interleaved row (1024 x 16B) into LDS
    for (int j = tid; j < TBL_FLOATS / 4; j += 256)
        async_cp16(gtbl + 4 * j, tbl + 4 * j);
    for (int j = tid; j < (2 * GRIDN) / 4; j += 256)
        async_cp16(rowg + 4 * j, Xint + 4 * j);
    async_wait0();
    __syncthreads();
    // de-interleave row: x[n1 + 64*n2] -> Xre/Xim[n2][n1]
    for (int j = tid; j < GRIDN; j += 256) {
        int n1 = j & 63, n2 = j >> 6;
        Xre[n2 * 68 + n1] = Xint[2 * j];
        Xim[n2 * 68 + n1] = Xint[2 * j + 1];
    }
#else
    for (int j = tid; j < TBL_FLOATS; j += 256) tbl[j] = gtbl[j];
    for (int j = tid; j < GRIDN; j += 256) {
        int n1 = j & 63, n2 = j >> 6;
        Xre[n2 * 68 + n1] = rowg[2 * j];
        Xim[n2 * 68 + n1] = rowg[2 * j + 1];
    }
#endif
    __syncthreads();

    const float* D32re = tbl + OFF_D32RE;
    const float* D32im = tbl + OFF_D32IM;
    const float* D32ni = tbl + OFF_D32NI;
    const float* D64re = tbl + OFF_D64RE;
    const float* D64im = tbl + OFF_D64IM;
    const float* D64ni = tbl + OFF_D64NI;
    const float* TWre  = tbl + OFF_TWRE;
    const float* TWim  = tbl + OFF_TWIM;

    // step 1: Z(32x64) = D32(32x32) * X(32x64)   [2x4 tiles, one per wave]
    cmatmul_tile(D32re, D32im, D32ni, D32_PITCH, Xre, Xim, 68, Zre, Zim, 68,
                 (wave >> 2) * 16, (wave & 3) * 16, 32, lane);
    __syncthreads();

    // step 2: twiddle + transpose: T[n1][k2] = Z[k2][n1] * W2048^(n1*k2)
    for (int j = tid; j < GRIDN; j += 256) {
        int n1 = j >> 5, k2 = j & 31;
        float zr = Zre[k2 * 68 + n1], zi = Zim[k2 * 68 + n1];
        float cs = TWre[n1 * TW_PITCH + k2], sn = TWim[n1 * TW_PITCH + k2];
        Tre[n1 * 36 + k2] = zr * cs - zi * sn;
        Tim[n1 * 36 + k2] = zr * sn + zi * cs;
    }
    __syncthreads();

    // step 3: O(64x32) = D64(64x64) * T(64x32)   [4x2 tiles, one per wave]
    cmatmul_tile(D64re, D64im, D64ni, D64_PITCH, Tre, Tim, 36, Ore, Oim, 36,
                 (wave >> 1) * 16, (wave & 1) * 16, 64, lane);
    __syncthreads();

    // store transposed: X_out[k] with k = 32*k1 + k2  ->  dst[c][k][row]
    float2* dp = dst + (size_t)c * GRIDN * GRIDN;
    for (int j = tid; j < GRIDN; j += 256) {
        float2 v;
        v.x = Ore[(j >> 5) * 36 + (j & 31)];
        v.y = Oim[(j >> 5) * 36 + (j & 31)];
        dp[(size_t)j * GRIDN + row] = v;
    }
}

// ---------------- Bessel I0 (Abramowitz-Stegun) ----------------
__device__ __forceinline__ float bessel_i0f(float x) {
    float ax = fabsf(x);
    if (ax < 3.75f) {
        float t = ax / 3.75f; t *= t;
        return 1.f + t * (3.5156229f + t * (3.0899424f + t * (1.2067492f +
               t * (0.2659732f + t * (0.0360768f + t * 0.0045813f)))));
    }
    float t = 3.75f / ax;
    return (expf(ax) * __frsqrt_rn(ax)) *
           (0.39894228f + t * (0.01328592f + t * (0.00225319f + t * (-0.00157565f +
            t * (0.00916281f + t * (-0.02057706f + t * (0.02635537f +
            t * (-0.01647633f + t * 0.00392377f))))))));
}

__device__ __forceinline__ void kb_weights(float g, int* idx, float* w) {
    int base = (int)floorf(g);
#pragma unroll
    for (int j = 0; j < KW; ++j) {
        int id = base + j - (KW / 2 - 1);
        float d = g - (float)id;
        float r = 2.f * d / (float)KW;
        float t = 1.f - r * r;
        w[j] = (t > 0.f) ? bessel_i0f(BETA_F * sqrtf(t)) : 0.f;
        id %= GRIDN; if (id < 0) id += GRIDN;
        idx[j] = id;
    }
}

// ---------------- kernel 3: KB degrid (6x6 gather per visibility) ----------
// Whole 134 MB spectrum fits in the 192 MB L2, so gathers are L2-resident.
__global__ void nufft_degrid(const float2* __restrict__ Kgrid,
                             const float* __restrict__ uu,
                             const float* __restrict__ vv,
                             float* __restrict__ out) {
    int n = blockIdx.x * blockDim.x + threadIdx.x;
    if (n >= NVIS) return;
    float gu = fmodf(uu[n] * GSCALE, (float)GRIDN); if (gu < 0.f) gu += (float)GRIDN;
    float gv = fmodf(vv[n] * GSCALE, (float)GRIDN); if (gv < 0.f) gv += (float)GRIDN;
    int   iu[KW], iv[KW];
    float wu[KW], wv[KW];
    kb_weights(gu, iu, wu);
    kb_weights(gv, iv, wv);

    float accRe[NCHAN] = {0.f}, accIm[NCHAN] = {0.f};
#pragma unroll
    for (int i = 0; i < KW; ++i) {
        float rRe[NCHAN] = {0.f}, rIm[NCHAN] = {0.f};
        const float2* rowp = Kgrid + (size_t)iv[i] * GRIDN;
#pragma unroll
        for (int j = 0; j < KW; ++j) {
            float w = wu[j];
#pragma unroll
            for (int c = 0; c < NCHAN; ++c) {
                float2 val = rowp[(size_t)c * GRIDN * GRIDN + iu[j]];
                rRe[c] += w * val.x;
                rIm[c] += w * val.y;
            }
        }
#pragma unroll
        for (int c = 0; c < NCHAN; ++c) {
            accRe[c] += wv[i] * rRe[c];
            accIm[c] += wv[i] * rIm[c];
        }
    }
#pragma unroll
    for (int c = 0; c < NCHAN; ++c) {
        size_t o = ((size_t)c * NVIS + n) * 2;
        out[o]     = accRe[c];
        out[o + 1] = accIm[c];
    }
}

// ---------------- launcher ----------------
extern "C" void kernel_launch(void* const* d_in, const int* in_sizes, int n_in,
                              void* d_out, int out_size, void* d_ws, size_t ws_size,
                              hipStream_t stream) {
    const float* img = (const float*)d_in[0];
    const float* uu  = (const float*)d_in[1];
    const float* vv  = (const float*)d_in[2];
    float* out = (float*)d_out;

    const size_t gridElems = (size_t)NCHAN * GRIDN * GRIDN;
    float2* bufA = (float2*)d_ws;
    float2* bufB = bufA + gridElems;
    float*  tbl  = (float*)(bufB + gridElems);   // 84.5 KB table region

    // 0) DFT matrices + twiddles, computed once (L2-resident thereafter)
    nufft_make_tables<<<(D64_PLANE + 255) / 256, 256, 0, stream>>>(tbl);

    // 1) apodization-corrected, zero-padded, rolled grid
    {
        int blocks = (int)((gridElems + 255) / 256);
        nufft_prep<<<blocks, 256, 0, stream>>>(img, bufA);
    }
    // 2) 2D FFT = row FFTs (write transposed) twice
    {
        dim3 g(GRIDN, NCHAN);
        nufft_fft2048<<<g, 256, 0, stream>>>((const float2*)bufA, bufB, tbl);
        nufft_fft2048<<<g, 256, 0, stream>>>((const float2*)bufB, bufA, tbl);
    }
    // 3) Kaiser-Bessel degrid
    {
        int blocks = (NVIS + 127) / 128;
        nufft_degrid<<<blocks, 128, 0, stream>>>((const float2*)bufA, uu, vv, out);
    }
}